// FreeYOLOvx_50379966382495
// MI455X (gfx1250) — compile-verified
//
#include <hip/hip_runtime.h>

// ---------------- configuration ----------------
#define NB        8192          // 13-bit radix histogram buckets
#define CAND_MAX  4096          // candidate cap per level (expected ~1500)
#define TOPK_N    1000
#define NCLS      80

// level sizes (2560x2560 input): M = H*W, N = M*C
static const long long N_LVL[3] = { 8192000LL, 2048000LL, 512000LL };
static const int       W_LVL[3] = { 320, 160, 80 };
static const float     S_LVL[3] = { 8.f, 16.f, 32.f };

// ---------------- CDNA5 async global->LDS path (guarded probes) ----------------
#if defined(__has_builtin)
# if __has_builtin(__builtin_amdgcn_global_load_async_to_lds_b128) && \
     __has_builtin(__builtin_amdgcn_s_wait_asynccnt)
#  define USE_ASYNC_LDS 1
# endif
# if __has_builtin(__builtin_amdgcn_s_wait_tensorcnt)
#  define HAVE_TENSORCNT 1
# endif
#endif

typedef int v4i __attribute__((ext_vector_type(4)));
typedef __attribute__((address_space(1))) v4i gv4i;   // global (prints as __device__ *)
typedef __attribute__((address_space(3))) v4i lv4i;   // LDS    (prints as __shared__ *)

__device__ __forceinline__ void async_copy16(const float4* g, float4* lds) {
#ifdef USE_ASYNC_LDS
  // generic LDS pointer: low 32 bits are the LDS byte offset (aperture in high bits)
  __builtin_amdgcn_global_load_async_to_lds_b128(
      (gv4i*)(unsigned long long)g,
      (lv4i*)(unsigned)(unsigned long long)lds,
      0, 0);
#else
  *lds = *g;
#endif
}

template <int N>
__device__ __forceinline__ void async_wait() {
#ifdef USE_ASYNC_LDS
  __builtin_amdgcn_s_wait_asynccnt(N);
#endif
}

// order-preserving float-bits -> unsigned key (monotonic: larger float -> larger key)
__device__ __forceinline__ unsigned f2key(unsigned u) {
  return u ^ ((unsigned)((int)u >> 31) | 0x80000000u);
}
__device__ __forceinline__ unsigned key2f(unsigned k) {
  return k ^ ((unsigned)((int)(~k) >> 31) | 0x80000000u);
}

// ---------------- kernel 0: zero workspace counters/histograms ----------------
__global__ void zero_ws_kernel(unsigned* __restrict__ hist, unsigned* __restrict__ cnt) {
  int i = blockIdx.x * 256 + threadIdx.x;
  if (i < 3 * NB) hist[i] = 0u;
  if (i < 3)      cnt[i]  = 0u;
}

// ---------------- kernel 1: radix histogram with async LDS tile staging ----------------
__global__ __launch_bounds__(256)
void hist_kernel(const float* __restrict__ cls, long long n, unsigned* __restrict__ ghist) {
  __shared__ unsigned lhist[NB];
  __shared__ float4   tile[2][256];
  const int tid = threadIdx.x;

  for (int i = tid; i < NB; i += 256) lhist[i] = 0u;
  __syncthreads();

  const float4*   g4    = (const float4*)cls;
  const long long e4    = n >> 2;                 // exact multiple of 4
  const long long tiles = (e4 + 255) >> 8;        // exact multiple of 256 for all levels
  const long long gstep = (long long)gridDim.x;

  int p = 0;
  long long t = blockIdx.x;
  if (t < tiles) {
    long long idx = (t << 8) + tid;
    if (idx < e4) async_copy16(g4 + idx, &tile[0][tid]);
  }
  for (; t < tiles; t += gstep) {
    long long tn = t + gstep;                     // uniform per block
    if (tn < tiles) {
      long long idxn = (tn << 8) + tid;
      if (idxn < e4) async_copy16(g4 + idxn, &tile[p ^ 1][tid]);
      async_wait<1>();                            // current tile done (in-order completion)
    } else {
      async_wait<0>();
    }
    long long idx = (t << 8) + tid;
    if (idx < e4) {
      float4 v = tile[p][tid];
      atomicAdd(&lhist[f2key(__float_as_uint(v.x)) >> 19], 1u);
      atomicAdd(&lhist[f2key(__float_as_uint(v.y)) >> 19], 1u);
      atomicAdd(&lhist[f2key(__float_as_uint(v.z)) >> 19], 1u);
      atomicAdd(&lhist[f2key(__float_as_uint(v.w)) >> 19], 1u);
    }
    p ^= 1;
  }
  __syncthreads();
  for (int i = tid; i < NB; i += 256) {
    unsigned c = lhist[i];
    if (c) atomicAdd(&ghist[i], c);
  }
#ifdef HAVE_TENSORCNT
  __builtin_amdgcn_s_wait_tensorcnt(0);
#endif
}

// ---------------- kernel 2: find cutoff bucket (suffix count >= TOPK) ----------------
__global__ void cutoff_kernel(const unsigned* __restrict__ hist, unsigned* __restrict__ cut) {
  if (threadIdx.x != 0) return;
  const unsigned* h = hist + (long long)blockIdx.x * NB;
  unsigned acc = 0;
  int b = NB - 1;
  for (; b > 0; --b) {
    acc += h[b];
    if (acc >= TOPK_N) break;
  }
  cut[blockIdx.x] = (unsigned)b;
}

// ---------------- kernel 3: compact candidates >= cutoff bucket ----------------
__device__ __forceinline__ void push_cand(unsigned* cnt, unsigned long long* cand,
                                          unsigned key, unsigned idx) {
  unsigned pos = atomicAdd(cnt, 1u);
  if (pos < CAND_MAX)
    cand[pos] = ((unsigned long long)key << 32) | (unsigned)(~idx); // tie-break: low idx first
}

__global__ __launch_bounds__(256)
void compact_kernel(const float* __restrict__ cls, long long n,
                    const unsigned* __restrict__ cutp,
                    unsigned* __restrict__ cnt,
                    unsigned long long* __restrict__ cand) {
  const unsigned B = *cutp;                       // uniform scalar load
  const float4* g4 = (const float4*)cls;
  const long long e4 = n >> 2;
  const long long step = (long long)gridDim.x * 256;
  for (long long i = (long long)blockIdx.x * 256 + threadIdx.x; i < e4; i += step) {
    if (i + step < e4) __builtin_prefetch(g4 + i + step, 0, 1);  // global_prefetch_b8
    float4 v = g4[i];
    unsigned base = (unsigned)(i << 2);
    unsigned k;
    k = f2key(__float_as_uint(v.x)); if ((k >> 19) >= B) push_cand(cnt, cand, k, base + 0);
    k = f2key(__float_as_uint(v.y)); if ((k >> 19) >= B) push_cand(cnt, cand, k, base + 1);
    k = f2key(__float_as_uint(v.z)); if ((k >> 19) >= B) push_cand(cnt, cand, k, base + 2);
    k = f2key(__float_as_uint(v.w)); if ((k >> 19) >= B) push_cand(cnt, cand, k, base + 3);
  }
}

// ---------------- kernel 4: bitonic sort (desc) + decode + write outputs ----------------
__global__ __launch_bounds__(1024)
void sort_output_kernel(const unsigned* __restrict__ cntArr,
                        const unsigned long long* __restrict__ candAll,
                        const float* __restrict__ reg0,
                        const float* __restrict__ reg1,
                        const float* __restrict__ reg2,
                        float* __restrict__ out) {
  __shared__ unsigned long long s[CAND_MAX];      // 32 KB
  const int lvl = blockIdx.x;
  const int tid = threadIdx.x;

  unsigned cnt = cntArr[lvl];
  if (cnt > CAND_MAX) cnt = CAND_MAX;
  const unsigned long long* cand = candAll + (long long)lvl * CAND_MAX;

  for (int i = tid; i < CAND_MAX; i += 1024) s[i] = (i < (int)cnt) ? cand[i] : 0ull;
  __syncthreads();

  // bitonic sort, descending (largest key first; equal keys -> larger ~idx == smaller idx first)
  for (int k = 2; k <= CAND_MAX; k <<= 1) {
    for (int j = k >> 1; j > 0; j >>= 1) {
      for (int i = tid; i < CAND_MAX; i += 1024) {
        int ixj = i ^ j;
        if (ixj > i) {
          unsigned long long a = s[i], b = s[ixj];
          bool desc = ((i & k) == 0);
          if (desc ? (a < b) : (a > b)) { s[i] = b; s[ixj] = a; }
        }
      }
      __syncthreads();
    }
  }

  if (tid < TOPK_N) {
    const int pos = tid;
    unsigned long long v = s[pos];
    unsigned key = (unsigned)(v >> 32);
    unsigned idx = ~((unsigned)v);
    float x = __uint_as_float(key2f(key));        // bit-exact original logit
    float score = 1.0f / (1.0f + expf(-x));
    bool keep = (pos < (int)cnt) && (score > 0.05f);

    float b0 = 0.f, b1 = 0.f, b2 = 0.f, b3 = 0.f, sc = 0.f;
    int lab = -1;
    if (keep) {
      const float* regp = (lvl == 0) ? reg0 : ((lvl == 1) ? reg1 : reg2);
      const int   w  = W_LVL[lvl];
      const float st = S_LVL[lvl];
      unsigned anchor = idx / (unsigned)NCLS;
      lab = (int)(idx - anchor * (unsigned)NCLS);
      float r0 = regp[anchor * 4 + 0];
      float r1 = regp[anchor * 4 + 1];
      float r2 = regp[anchor * 4 + 2];
      float r3 = regp[anchor * 4 + 3];
      float ax = ((float)(anchor % (unsigned)w) + 0.5f) * st;
      float ay = ((float)(anchor / (unsigned)w) + 0.5f) * st;
      float cx = ax + r0 * st;
      float cy = ay + r1 * st;
      float bw = expf(r2) * st;
      float bh = expf(r3) * st;
      b0 = cx - 0.5f * bw; b1 = cy - 0.5f * bh;
      b2 = cx + 0.5f * bw; b3 = cy + 0.5f * bh;
      sc = score;
    }
    const int o = lvl * TOPK_N + pos;
    out[o * 4 + 0] = b0;
    out[o * 4 + 1] = b1;
    out[o * 4 + 2] = b2;
    out[o * 4 + 3] = b3;
    out[3 * TOPK_N * 4 + o] = sc;                       // scores at [12000, 15000)
    ((int*)out)[3 * TOPK_N * 4 + 3 * TOPK_N + o] = lab; // labels (raw int32) at [15000, 18000)
  }
}

// ---------------- host launcher ----------------
extern "C" void kernel_launch(void* const* d_in, const int* in_sizes, int n_in,
                              void* d_out, int out_size, void* d_ws, size_t ws_size,
                              hipStream_t stream) {
  (void)in_sizes; (void)n_in; (void)out_size; (void)ws_size;
  const float* cls[3] = { (const float*)d_in[0], (const float*)d_in[2], (const float*)d_in[4] };
  const float* reg0 = (const float*)d_in[1];
  const float* reg1 = (const float*)d_in[3];
  const float* reg2 = (const float*)d_in[5];
  float* out = (float*)d_out;

  // workspace layout
  char* ws = (char*)d_ws;
  unsigned* hist = (unsigned*)ws;                                   // 3*NB u32 = 98304 B
  unsigned* cut  = (unsigned*)(ws + 3 * NB * sizeof(unsigned));     // 3 u32
  unsigned* cnt  = cut + 3;                                         // 3 u32
  unsigned long long* cand =
      (unsigned long long*)(ws + ((3 * NB + 8) * sizeof(unsigned) + 7) / 8 * 8); // 3*CAND_MAX u64

  zero_ws_kernel<<<(3 * NB + 255) / 256, 256, 0, stream>>>(hist, cnt);

  const int histGrid[3] = { 2048, 2000, 500 };   // == #tiles clamped
  for (int l = 0; l < 3; ++l)
    hist_kernel<<<histGrid[l], 256, 0, stream>>>(cls[l], N_LVL[l], hist + l * NB);

  cutoff_kernel<<<3, 32, 0, stream>>>(hist, cut);

  const int compGrid[3] = { 1024, 512, 256 };
  for (int l = 0; l < 3; ++l)
    compact_kernel<<<compGrid[l], 256, 0, stream>>>(cls[l], N_LVL[l],
                                                    cut + l, cnt + l,
                                                    cand + (long long)l * CAND_MAX);

  sort_output_kernel<<<3, 1024, 0, stream>>>(cnt, cand, reg0, reg1, reg2, out);
}